// AGNN_SAT_51470888075465
// MI455X (gfx1250) — compile-verified
//
#include <hip/hip_runtime.h>
#include <hip/hip_bf16.h>

typedef float v2f __attribute__((ext_vector_type(2)));
typedef float v8f __attribute__((ext_vector_type(8)));
typedef int idx_t;   // JAX default x64-disabled => int32 indices

#define HDIM 256
#define HEADW 128
#define COUT 6

__device__ __forceinline__ void atomAddF(float* p, float v) {
  __hip_atomic_fetch_add(p, v, __ATOMIC_RELAXED, __HIP_MEMORY_SCOPE_AGENT);
}

// ---------------- utility fills ----------------
__global__ void fill_f32(float* __restrict__ p, float v, long n) {
  long i = (long)blockIdx.x * blockDim.x + threadIdx.x;
  long stride = (long)gridDim.x * blockDim.x;
  for (; i < n; i += stride) p[i] = v;
}

// ---------------- degree / norm ----------------
__global__ void deg_accum(const idx_t* __restrict__ dst, float* __restrict__ deg, int E) {
  int e = blockIdx.x * blockDim.x + threadIdx.x;
  if (e < E) atomAddF(&deg[dst[e]], 1.0f);
}

__global__ void rsqrt_ip(float* __restrict__ d, int n) {
  int i = blockIdx.x * blockDim.x + threadIdx.x;
  if (i < n) d[i] = rsqrtf(d[i]);
}

// ---------------- layer 1 (rank-1 shortcut) ----------------
// s[n] = sum_{e: dst==n} dinv[src]*dinv[n]*x[src]
__global__ void l1_edge(const idx_t* __restrict__ src, const idx_t* __restrict__ dst,
                        const float* __restrict__ dinv, const float* __restrict__ x,
                        float* __restrict__ s, int E) {
  int e = blockIdx.x * blockDim.x + threadIdx.x;
  if (e < E) {
    int a = src[e], b = dst[e];
    atomAddF(&s[b], dinv[a] * dinv[b] * x[a]);
  }
}

// h1[n,c] = relu((s[n] + dinv[n]^2 * x[n]) * W1[c] + b1[c])
__global__ void __launch_bounds__(HDIM) l1_node(const float* __restrict__ s,
                                                const float* __restrict__ dinv,
                                                const float* __restrict__ x,
                                                const float* __restrict__ W1,
                                                const float* __restrict__ b1,
                                                float* __restrict__ h) {
  int n = blockIdx.x, c = threadIdx.x;
  float di = dinv[n];
  float t = (s[n] + di * di * x[n]) * W1[c] + b1[c];
  h[(size_t)n * HDIM + c] = fmaxf(t, 0.0f);
}

// ---------------- dense GEMM via V_WMMA_F32_16X16X4_F32 ----------------
// Out[nrows, 256] = A[nrows, 256] @ W[256, 256], fp32.
// Block = 256 threads = 8 waves; wave w computes a 16x32 tile (cols [32w,32w+32)).
// nrows % 16 == 0 (100000 = 6250*16) so EXEC is all-ones everywhere (WMMA req).
__global__ void __launch_bounds__(256) gemm_wmma_f32(const float* __restrict__ A,
                                                     const float* __restrict__ W,
                                                     float* __restrict__ Out) {
  const int wave = threadIdx.x >> 5;
  const int lane = threadIdx.x & 31;
  const int hi   = lane >> 4;     // lanes 16-31 hold K-pairs {2,3}
  const int lo   = lane & 15;
  const int m0   = blockIdx.x * 16;
  const int n0   = wave * 32;

  const float* __restrict__ arow = A + (size_t)(m0 + lo) * HDIM;
  v8f c0 = {}; v8f c1 = {};

#pragma unroll 4
  for (int k = 0; k < HDIM; k += 4) {
    const int ka = k + 2 * hi;           // A: VGPR0->K=ka, VGPR1->K=ka+1
    v2f a; a.x = arow[ka]; a.y = arow[ka + 1];
    const float* __restrict__ w0 = W + (size_t)ka * HDIM;
    v2f b0; b0.x = w0[n0 + lo];       b0.y = w0[HDIM + n0 + lo];
    v2f b1; b1.x = w0[n0 + 16 + lo];  b1.y = w0[HDIM + n0 + 16 + lo];
    c0 = __builtin_amdgcn_wmma_f32_16x16x4_f32(false, a, false, b0, (short)0, c0, false, false);
    c1 = __builtin_amdgcn_wmma_f32_16x16x4_f32(false, a, false, b1, (short)0, c1, false, false);
  }

  // C/D layout: VGPR r -> row m0+r (lanes 0-15) / m0+8+r (lanes 16-31), col n-part = lo
  float* __restrict__ orow = Out + (size_t)(m0 + hi * 8) * HDIM + n0 + lo;
#pragma unroll
  for (int r = 0; r < 8; ++r) {
    orow[(size_t)r * HDIM]      = c0[r];
    orow[(size_t)r * HDIM + 16] = c1[r];
  }
}

// ---------------- edge scatter: agg[dst] += norm_e * hw[src] ----------------
// One wave32 per edge; 256 channels via 2 x (32 lanes x float4).
__global__ void __launch_bounds__(256) edge_scatter(const idx_t* __restrict__ src,
                                                    const idx_t* __restrict__ dst,
                                                    const float* __restrict__ dinv,
                                                    const float* __restrict__ hw,
                                                    float* __restrict__ agg, int E) {
  int wv = blockIdx.x * (blockDim.x >> 5) + (threadIdx.x >> 5);
  if (wv >= E) return;
  int lane = threadIdx.x & 31;
  int s = src[wv], d = dst[wv];
  float nrm = dinv[s] * dinv[d];
  const float4* __restrict__ in = (const float4*)(hw + (size_t)s * HDIM);
  float* __restrict__ out = agg + (size_t)d * HDIM;
#pragma unroll
  for (int it = 0; it < 2; ++it) {
    int c4 = it * 32 + lane;
    float4 v = in[c4];
    int c = c4 * 4;
    atomAddF(out + c + 0, nrm * v.x);
    atomAddF(out + c + 1, nrm * v.y);
    atomAddF(out + c + 2, nrm * v.z);
    atomAddF(out + c + 3, nrm * v.w);
  }
}

// h_next[n,c] = relu(agg[n,c] + dinv[n]^2 * hw[n,c] + b[c])   (self-loop folded in)
__global__ void __launch_bounds__(HDIM) layer_epilogue(const float* __restrict__ agg,
                                                       const float* __restrict__ hw,
                                                       const float* __restrict__ dinv,
                                                       const float* __restrict__ bias,
                                                       float* __restrict__ hout) {
  int n = blockIdx.x, c = threadIdx.x;
  float di = dinv[n];
  size_t i = (size_t)n * HDIM + c;
  hout[i] = fmaxf(agg[i] + di * di * hw[i] + bias[c], 0.0f);
}

// ---------------- mean pool ----------------
__global__ void __launch_bounds__(HDIM) pool_accum(const float* __restrict__ h,
                                                   const idx_t* __restrict__ batch,
                                                   float* __restrict__ sums,
                                                   float* __restrict__ cnt) {
  int n = blockIdx.x, c = threadIdx.x;
  int b = batch[n];
  atomAddF(&sums[(size_t)b * HDIM + c], h[(size_t)n * HDIM + c]);
  if (c == 0) atomAddF(&cnt[b], 1.0f);
}

// ---------------- head MLP: one block per graph ----------------
__global__ void __launch_bounds__(HEADW) head_mlp(const float* __restrict__ sums,
                                                  const float* __restrict__ cnt,
                                                  const float* __restrict__ svm,
                                                  const float* __restrict__ Wf1,
                                                  const float* __restrict__ bf1,
                                                  const float* __restrict__ Wf2,
                                                  const float* __restrict__ bf2,
                                                  float* __restrict__ out) {
  __shared__ float feat[HDIM + 1];
  __shared__ float t1[HEADW];
  int b = blockIdx.x, j = threadIdx.x;
  float inv = 1.0f / fmaxf(cnt[b], 1.0f);
  feat[j]          = sums[(size_t)b * HDIM + j] * inv;
  feat[j + HEADW]  = sums[(size_t)b * HDIM + j + HEADW] * inv;
  if (j == 0) feat[HDIM] = svm[b];
  __syncthreads();
  float acc = bf1[j];
  for (int k = 0; k < HDIM + 1; ++k) acc = fmaf(feat[k], Wf1[(size_t)k * HEADW + j], acc);
  t1[j] = fmaxf(acc, 0.0f);
  __syncthreads();
  if (j < COUT) {
    float o = bf2[j];
    for (int k = 0; k < HEADW; ++k) o = fmaf(t1[k], Wf2[(size_t)k * COUT + j], o);
    out[(size_t)b * COUT + j] = o;
  }
}

// ---------------- orchestration ----------------
extern "C" void kernel_launch(void* const* d_in, const int* in_sizes, int n_in,
                              void* d_out, int out_size, void* d_ws, size_t ws_size,
                              hipStream_t stream) {
  const float* x        = (const float*)d_in[0];
  const idx_t* ei       = (const idx_t*)d_in[1];   // [2,E]: row0=src, row1=dst
  const idx_t* batch    = (const idx_t*)d_in[2];
  const float* svm      = (const float*)d_in[3];
  const float* W1 = (const float*)d_in[4];  const float* b1 = (const float*)d_in[5];
  const float* W2 = (const float*)d_in[6];  const float* b2 = (const float*)d_in[7];
  const float* W3 = (const float*)d_in[8];  const float* b3 = (const float*)d_in[9];
  const float* Wf1 = (const float*)d_in[10]; const float* bf1 = (const float*)d_in[11];
  const float* Wf2 = (const float*)d_in[12]; const float* bf2 = (const float*)d_in[13];
  float* out = (float*)d_out;

  const int N = in_sizes[0];        // C_IN == 1
  const int E = in_sizes[1] / 2;
  const int B = in_sizes[3];
  const idx_t* src = ei;
  const idx_t* dst = ei + E;

  // workspace carve-up (256B aligned)
  auto align256 = [](size_t v) { return (v + 255) & ~(size_t)255; };
  char* ws = (char*)d_ws;
  size_t off = 0;
  float* dinv = (float*)(ws + off); off = align256(off + (size_t)N * 4);
  float* s1   = (float*)(ws + off); off = align256(off + (size_t)N * 4);
  float* sums = (float*)(ws + off); off = align256(off + (size_t)B * HDIM * 4);
  float* cnt  = (float*)(ws + off); off = align256(off + (size_t)B * 4);
  float* bufA = (float*)(ws + off); off = align256(off + (size_t)N * HDIM * 4);
  float* bufB = (float*)(ws + off); off = align256(off + (size_t)N * HDIM * 4);
  float* bufC = (float*)(ws + off); off = align256(off + (size_t)N * HDIM * 4);

  const int TB = 256;
  const long NH = (long)N * HDIM;
  const int fillGrid = 2048;

  // deg = 1 (self-loop), then += indegree; dinv = rsqrt(deg)
  fill_f32<<<fillGrid, TB, 0, stream>>>(dinv, 1.0f, N);
  fill_f32<<<fillGrid, TB, 0, stream>>>(s1, 0.0f, N);
  fill_f32<<<fillGrid, TB, 0, stream>>>(sums, 0.0f, (long)B * HDIM);
  fill_f32<<<fillGrid, TB, 0, stream>>>(cnt, 0.0f, B);
  deg_accum<<<(E + TB - 1) / TB, TB, 0, stream>>>(dst, dinv, E);
  rsqrt_ip<<<(N + TB - 1) / TB, TB, 0, stream>>>(dinv, N);

  // layer 1 (rank-1): scalar edge reduce, then outer product with W1
  l1_edge<<<(E + TB - 1) / TB, TB, 0, stream>>>(src, dst, dinv, x, s1, E);
  l1_node<<<N, HDIM, 0, stream>>>(s1, dinv, x, W1, b1, bufA);

  // layers 2 & 3: WMMA GEMM -> zero agg -> edge scatter -> epilogue(relu+bias+selfloop)
  const int gemmGrid = N / 16;                 // 100000/16 = 6250, exact
  const int scatGrid = (E + 7) / 8;            // 8 waves per block, 1 wave per edge

  gemm_wmma_f32<<<gemmGrid, 256, 0, stream>>>(bufA, W2, bufB);
  fill_f32<<<fillGrid, TB, 0, stream>>>(bufC, 0.0f, NH);
  edge_scatter<<<scatGrid, 256, 0, stream>>>(src, dst, dinv, bufB, bufC, E);
  layer_epilogue<<<N, HDIM, 0, stream>>>(bufC, bufB, dinv, b2, bufA);

  gemm_wmma_f32<<<gemmGrid, 256, 0, stream>>>(bufA, W3, bufB);
  fill_f32<<<fillGrid, TB, 0, stream>>>(bufC, 0.0f, NH);
  edge_scatter<<<scatGrid, 256, 0, stream>>>(src, dst, dinv, bufB, bufC, E);
  layer_epilogue<<<N, HDIM, 0, stream>>>(bufC, bufB, dinv, b3, bufA);

  // mean pool + head
  pool_accum<<<N, HDIM, 0, stream>>>(bufA, batch, sums, cnt);
  head_mlp<<<B, HEADW, 0, stream>>>(sums, cnt, svm, Wf1, bf1, Wf2, bf2, out);
}